// NelectronVanderWithMult_11699490915045
// MI455X (gfx1250) — compile-verified
//
#include <hip/hip_runtime.h>

typedef __attribute__((ext_vector_type(2))) float v2f;
typedef __attribute__((ext_vector_type(8))) float v8f;

#define DIM 64
#define WAVES_PER_BLOCK 8

// One wave32 per row of x[65536, 64].
// Pairwise differences e_i - e_j built as a rank-2 WMMA (V_WMMA_F32_16X16X4_F32):
//   A(16x4) = [u_m, 1, u_m, 1]   (lanes 16-31 hold K=2,3 -> same data, no masking)
//   B(4x16) = [1; -v_n; 0; 0]    (K=2,3 rows zeroed -> their A contribution is *0)
//   D[m,n]  = u_m - v_n          (exact fp32: products by +/-1/0)
// Tiles ti<=tj only; diagonal tiles mask m>=n to 1.0 (log contribution 0).
// Adjacent differences are multiplied pairwise so each v_log_f32 covers two
// pair terms: log2|a*b| = log2|a| + log2|b| (safe: |diff| >= ~5e-10 if nonzero).
__global__ __launch_bounds__(256) void vander_wmma_kernel(
    const float* __restrict__ x,
    const float* __restrict__ alpha_p,
    const float* __restrict__ beta_p,
    float* __restrict__ out,
    int nrows) {
  __shared__ float smem[WAVES_PER_BLOCK * DIM];

  const int lane = threadIdx.x & 31;
  const int wave = threadIdx.x >> 5;
  const int row = blockIdx.x * WAVES_PER_BLOCK + wave;
  // Clamp for loads (keep all lanes converged); guard only the final store.
  const int rowc = (row < nrows) ? row : 0;

  const float alpha = alpha_p[0];
  const float beta  = beta_p[0];

  // Coalesced: wave reads one contiguous 256B row segment, 8B per lane.
  const float2 xv = ((const float2*)(x + (size_t)rowc * DIM))[lane];

  // e = exp(-alpha*x) = exp2(-alpha*log2(e)*x) -> v_exp_f32
  const float c = -alpha * 1.4426950408889634f;
  const float e0 = __builtin_amdgcn_exp2f(c * xv.x);
  const float e1 = __builtin_amdgcn_exp2f(c * xv.y);

  float* my = &smem[wave * DIM];
  my[2 * lane]     = e0;
  my[2 * lane + 1] = e1;
  __syncthreads();

  const bool lo   = (lane < 16);
  const int  l15  = lane & 15;
  const int  mofs = lo ? 0 : 8;          // row offset of this lane's D elements

  // Unconditional LDS gathers, hoisted: g[t] = e[16*t + l15] for ALL lanes.
  float g[4];
  #pragma unroll
  for (int t = 0; t < 4; ++t) g[t] = my[16 * t + l15];

  const float bmask = lo ? 1.0f : 0.0f;  // zeroes B's K=2,3 rows

  // Diagonal-tile element mask, hoisted (identical for all 4 diagonal tiles).
  bool keep[8];
  #pragma unroll
  for (int v = 0; v < 8; ++v) keep[v] = (v + mofs) < l15;

  // 4 accumulators to break the add chain; logs go down the TRANS pipe.
  float acc0 = 0.0f, acc1 = 0.0f, acc2 = 0.0f, acc3 = 0.0f;

  #pragma unroll
  for (int ti = 0; ti < 4; ++ti) {
    v2f A;
    A.x = g[ti];   // A[m,0] (lanes 0-15) and A[m,2] (lanes 16-31)
    A.y = 1.0f;    // A[m,1]              and A[m,3]
    #pragma unroll
    for (int tj = ti; tj < 4; ++tj) {
      v2f B;
      B.x = bmask;           // B[0,n] = 1, B[2,n] = 0
      B.y = -g[tj] * bmask;  // B[1,n] = -v_n, B[3,n] = 0
      v8f Cz = {0.f, 0.f, 0.f, 0.f, 0.f, 0.f, 0.f, 0.f};
      v8f D = __builtin_amdgcn_wmma_f32_16x16x4_f32(
          false, A, false, B, (short)0, Cz, false, false);

      // D layout: VGPR v, lanes 0-15 -> (M=v, N=lane); lanes 16-31 -> (M=v+8, N=lane-16)
      #pragma unroll
      for (int p = 0; p < 4; ++p) {
        float d0 = D[2 * p];
        float d1 = D[2 * p + 1];
        if (ti == tj) {               // constant-folds away for off-diagonal tiles
          d0 = keep[2 * p]     ? d0 : 1.0f;
          d1 = keep[2 * p + 1] ? d1 : 1.0f;
        }
        // one v_mul + one v_log (|.| folds into the log input modifier)
        const float lg = __builtin_amdgcn_logf(fabsf(d0 * d1));
        if      (p == 0) acc0 += lg;
        else if (p == 1) acc1 += lg;
        else if (p == 2) acc2 += lg;
        else             acc3 += lg;
      }
    }
  }

  // Fused epilogue: single butterfly reduction of the combined per-lane value.
  // out = 2*(-beta*sum(x) + ln2 * sum(log2)); per-lane partials are additive.
  float part = -2.0f * beta * (xv.x + xv.y) +
               (2.0f * 0.6931471805599453f) * ((acc0 + acc1) + (acc2 + acc3));

  // wave32 butterfly via ds_swizzle_b32 (xor masks 16,8,4,2,1; and_mask=0x1f)
  part += __int_as_float(__builtin_amdgcn_ds_swizzle(__float_as_int(part), 0x401F));
  part += __int_as_float(__builtin_amdgcn_ds_swizzle(__float_as_int(part), 0x201F));
  part += __int_as_float(__builtin_amdgcn_ds_swizzle(__float_as_int(part), 0x101F));
  part += __int_as_float(__builtin_amdgcn_ds_swizzle(__float_as_int(part), 0x081F));
  part += __int_as_float(__builtin_amdgcn_ds_swizzle(__float_as_int(part), 0x041F));

  if (lane == 0 && row < nrows) {
    out[row] = part;
  }
}

extern "C" void kernel_launch(void* const* d_in, const int* in_sizes, int n_in,
                              void* d_out, int out_size, void* d_ws, size_t ws_size,
                              hipStream_t stream) {
  const float* x     = (const float*)d_in[0];
  const float* alpha = (const float*)d_in[1];
  const float* beta  = (const float*)d_in[2];
  float* out = (float*)d_out;

  const int nrows = in_sizes[0] / DIM;  // 65536
  const int blocks = (nrows + WAVES_PER_BLOCK - 1) / WAVES_PER_BLOCK;

  vander_wmma_kernel<<<blocks, 32 * WAVES_PER_BLOCK, 0, stream>>>(
      x, alpha, beta, out, nrows);
}